// NodeFormerConv_29781303230687
// MI455X (gfx1250) — compile-verified
//
#include <hip/hip_runtime.h>

// ---------------- problem constants ----------------
#define NN      50000
#define IN_DIM  128
#define DD      64
#define HH      4
#define MM      30
#define KGUM    8
#define EE      500000
#define QKV_DIM (DD*HH)          // 256
#define HM      (HH*MM)          // 120
#define CHUNK   512
#define NCHUNK  98               // ceil(NN/CHUNK)

#define DN_SCALE   0.35355339059327373f   // 64^(-1/4)
#define RATIO      0.18257418583505536f   // 1/sqrt(30)
#define QK_SCALE   2.0f                   // 1/sqrt(tau), tau=0.25
#define FEAT_EPS   1e-6f

typedef __attribute__((ext_vector_type(2))) float v2f;
typedef __attribute__((ext_vector_type(8))) float v8f;

static __device__ __forceinline__ v8f wmma4(v2f a, v2f b, v8f c) {
  // D = A(16x4 f32) * B(4x16 f32) + C(16x16 f32), wave32
  return __builtin_amdgcn_wmma_f32_16x16x4_f32(false, a, false, b, (short)0, c,
                                               false, false);
}

static __device__ __forceinline__ unsigned fkey(float f) {
  unsigned u = __float_as_uint(f);
  return (u & 0x80000000u) ? ~u : (u | 0x80000000u);
}
static __device__ __forceinline__ float funkey(unsigned u) {
  return (u & 0x80000000u) ? __uint_as_float(u & 0x7fffffffu)
                           : __uint_as_float(~u);
}

// ------------- gfx1250 async global->LDS staging (ASYNCcnt path) -------------
#if defined(__has_builtin)
#if __has_builtin(__builtin_amdgcn_global_load_async_to_lds_b128)
#define HAVE_ASYNC_LDS 1
#endif
#endif

#define AS1 __attribute__((address_space(1)))
#define AS3 __attribute__((address_space(3)))

typedef int vsi4 __attribute__((vector_size(16)));  // matches builtin param type

static __device__ __forceinline__ void async_cp16(const float* g, float* l) {
#ifdef HAVE_ASYNC_LDS
  // generic->AS1: bit-identical; generic LDS ptr low 32 bits = AS3 offset
  unsigned long long ga = (unsigned long long)g;
  unsigned la = (unsigned)(unsigned long long)l;
  __builtin_amdgcn_global_load_async_to_lds_b128((AS1 vsi4*)ga, (AS3 vsi4*)la,
                                                 0, 0);
#else
  l[0] = g[0]; l[1] = g[1]; l[2] = g[2]; l[3] = g[3];
#endif
}
static __device__ __forceinline__ void wait_async0() {
#ifdef HAVE_ASYNC_LDS
#if __has_builtin(__builtin_amdgcn_s_wait_asynccnt)
  __builtin_amdgcn_s_wait_asynccnt(0);
#else
  asm volatile("s_wait_asynccnt 0" ::: "memory");
#endif
#endif
}

// ============ K1: fused QKV GEMM  q/k/v = (z @ W^T + b) [* scale] ============
// grid.x = NN/16 row tiles, block = 256 (8 waves); each wave covers 6 of the
// 48 output col tiles (q|k|v = 768 cols) so z is streamed from HBM exactly once.
__global__ void __launch_bounds__(256) gemm_qkv_k(
    const float* __restrict__ z,
    const float* __restrict__ Wq, const float* __restrict__ bq,
    const float* __restrict__ Wk, const float* __restrict__ bk,
    const float* __restrict__ Wv, const float* __restrict__ bv,
    float* __restrict__ q, float* __restrict__ k, float* __restrict__ v) {
  __shared__ float zt[16][132];                 // stride 132 -> conflict-free
  const int r0  = blockIdx.x * 16;
  const int tid = threadIdx.x;

  // async stage: 16x128 floats = 512 x 16B chunks, 2 per thread
#pragma unroll
  for (int it = 0; it < 2; ++it) {
    int c = tid + it * 256;                     // [0,512)
    int rr = c >> 5, cc = (c & 31) * 4;
    async_cp16(z + (size_t)(r0 + rr) * IN_DIM + cc, &zt[rr][cc]);
  }
  wait_async0();
  __syncthreads();

  const int wave = tid >> 5, lane = tid & 31;
  const int l = lane & 15, half = lane >> 4;

  for (int t6 = 0; t6 < 6; ++t6) {
    const int tile = wave * 6 + t6;             // 0..47
    const int mat  = tile >> 4;                 // 0=q 1=k 2=v
    const int c0   = (tile & 15) * 16;
    const float* W    = (mat == 0) ? Wq : (mat == 1) ? Wk : Wv;
    const float* bias = (mat == 0) ? bq : (mat == 1) ? bk : bv;
    float*       out  = (mat == 0) ? q  : (mat == 1) ? k  : v;
    const float  scl  = (mat == 2) ? 1.0f : QK_SCALE;

    const float* wrow = W + (size_t)(c0 + l) * IN_DIM;
    v8f c = {0.f, 0.f, 0.f, 0.f, 0.f, 0.f, 0.f, 0.f};
#pragma unroll 8
    for (int kk = 0; kk < IN_DIM; kk += 4) {
      v2f a, b;
      a.x = zt[l][kk + 2 * half];
      a.y = zt[l][kk + 2 * half + 1];
      b.x = wrow[kk + 2 * half];
      b.y = wrow[kk + 2 * half + 1];
      c = wmma4(a, b, c);
    }
    const float bb = bias[c0 + l];
#pragma unroll
    for (int i = 0; i < 8; ++i)
      out[(size_t)(r0 + i + 8 * half) * QKV_DIM + c0 + l] = (c[i] + bb) * scl;
  }
}

// ============ K2: query features qp = ratio*(exp(dd - diag - rowmax)+eps) ====
__global__ void __launch_bounds__(256) qp_feat_k(
    const float* __restrict__ q, const float* __restrict__ proj,
    float* __restrict__ qp) {
  __shared__ float pj[MM][DD];
  for (int i = threadIdx.x; i < MM * DD; i += 256) pj[i / DD][i % DD] = proj[i];
  __syncthreads();
  int idx = blockIdx.x * 256 + threadIdx.x;
  if (idx >= NN * HH) return;
  int n = idx >> 2, h = idx & 3;
  const float* qrow = q + (size_t)n * QKV_DIM + h * DD;
  float data[DD];
  float diag = 0.f;
#pragma unroll
  for (int d = 0; d < DD; ++d) { float x = qrow[d] * DN_SCALE; data[d] = x; diag += x * x; }
  diag *= 0.5f;
  float* o = qp + (size_t)n * HM + h * MM;
  float mx = -3.4e38f;
  for (int m = 0; m < MM; ++m) {
    float s = 0.f;
#pragma unroll
    for (int d = 0; d < DD; ++d) s += data[d] * pj[m][d];
    o[m] = s;
    mx = fmaxf(mx, s);
  }
  for (int m = 0; m < MM; ++m)
    o[m] = RATIO * (expf(o[m] - diag - mx) + FEAT_EPS);
}

// ============ K3a: global per-head max of key dd ============
__global__ void __launch_bounds__(256) kmax_k(
    const float* __restrict__ kmat, const float* __restrict__ proj,
    unsigned* __restrict__ hmax) {
  __shared__ float pj[MM][DD];
  for (int i = threadIdx.x; i < MM * DD; i += 256) pj[i / DD][i % DD] = proj[i];
  __syncthreads();
  int idx = blockIdx.x * 256 + threadIdx.x;
  if (idx >= NN * HH) return;
  int n = idx >> 2, h = idx & 3;
  const float* krow = kmat + (size_t)n * QKV_DIM + h * DD;
  float data[DD];
#pragma unroll
  for (int d = 0; d < DD; ++d) data[d] = krow[d] * DN_SCALE;
  float mx = -3.4e38f;
  for (int m = 0; m < MM; ++m) {
    float s = 0.f;
#pragma unroll
    for (int d = 0; d < DD; ++d) s += data[d] * pj[m][d];
    mx = fmaxf(mx, s);
  }
  atomicMax(&hmax[h], fkey(mx));
}

// ============ K3b: key features with global max ============
__global__ void __launch_bounds__(256) kp_feat_k(
    const float* __restrict__ kmat, const float* __restrict__ proj,
    const unsigned* __restrict__ hmax, float* __restrict__ kp) {
  __shared__ float pj[MM][DD];
  for (int i = threadIdx.x; i < MM * DD; i += 256) pj[i / DD][i % DD] = proj[i];
  __syncthreads();
  int idx = blockIdx.x * 256 + threadIdx.x;
  if (idx >= NN * HH) return;
  int n = idx >> 2, h = idx & 3;
  const float mx = funkey(hmax[h]);
  const float* krow = kmat + (size_t)n * QKV_DIM + h * DD;
  float data[DD];
  float diag = 0.f;
#pragma unroll
  for (int d = 0; d < DD; ++d) { float x = krow[d] * DN_SCALE; data[d] = x; diag += x * x; }
  diag *= 0.5f;
  float* o = kp + (size_t)n * HM + h * MM;
  for (int m = 0; m < MM; ++m) {
    float s = 0.f;
#pragma unroll
    for (int d = 0; d < DD; ++d) s += data[d] * pj[m][d];
    o[m] = RATIO * (expf(s - diag - mx) + FEAT_EPS);
  }
}

// ============ K4: ks_tot[h,m] = sum_n kp[n,h,m] ============
__global__ void __launch_bounds__(256) kstot_k(const float* __restrict__ kp,
                                               float* __restrict__ ks_tot) {
  const int hm = blockIdx.x;  // 0..119
  float s = 0.f;
  for (int n = threadIdx.x; n < NN; n += 256) s += kp[(size_t)n * HM + hm];
  __shared__ float red[256];
  red[threadIdx.x] = s;
  __syncthreads();
  for (int st = 128; st > 0; st >>= 1) {
    if (threadIdx.x < st) red[threadIdx.x] += red[threadIdx.x + st];
    __syncthreads();
  }
  if (threadIdx.x == 0) ks_tot[hm] = red[0];
}

// ============ K5: chunked partials of kvs[h,k,m,d] and ks_sum[h,k,m] ========
// grid = (32 = h*8+k, NCHUNK), block = 256
__global__ void __launch_bounds__(256) kvs_part_k(
    const float* __restrict__ kp, const float* __restrict__ v,
    const float* __restrict__ gum, float* __restrict__ kvs_part,
    float* __restrict__ ks_part) {
  const int hk = blockIdx.x, h = hk >> 3, kg = hk & 7;
  const int n0 = blockIdx.y * CHUNK;
  const int n1 = (n0 + CHUNK < NN) ? (n0 + CHUNK) : NN;
  __shared__ float kps[64][31];
  __shared__ float gs[64];
  const int t = threadIdx.x;
  const int d = t & 63, mrow = t >> 6;  // mrow 0..3, m = mrow*8+j
  float acc[8] = {0.f, 0.f, 0.f, 0.f, 0.f, 0.f, 0.f, 0.f};
  float sks = 0.f;
  for (int nb = n0; nb < n1; nb += 64) {
    const int cnt = (n1 - nb < 64) ? (n1 - nb) : 64;
    __syncthreads();
    for (int i = t; i < cnt * MM; i += 256)
      kps[i / MM][i % MM] = kp[(size_t)(nb + i / MM) * HM + h * MM + i % MM];
    for (int i = t; i < cnt; i += 256)
      gs[i] = expf(gum[(size_t)(nb + i) * (HH * KGUM) + h * KGUM + kg]);
    __syncthreads();
    for (int nn = 0; nn < cnt; ++nn) {
      const float g = gs[nn];
      const float gv = g * v[(size_t)(nb + nn) * QKV_DIM + h * DD + d];
#pragma unroll
      for (int j = 0; j < 8; ++j) {
        int m = mrow * 8 + j;
        if (m < MM) acc[j] += kps[nn][m] * gv;
      }
      if (t < MM) sks += g * kps[nn][t];
    }
  }
  const size_t base = ((size_t)hk * NCHUNK + blockIdx.y) * (MM * DD);
#pragma unroll
  for (int j = 0; j < 8; ++j) {
    int m = mrow * 8 + j;
    if (m < MM) kvs_part[base + (size_t)m * DD + d] = acc[j];
  }
  if (t < MM) ks_part[((size_t)hk * NCHUNK + blockIdx.y) * MM + t] = sks;
}

// ============ K6: deterministic partial reduces ============
__global__ void __launch_bounds__(256) kvs_reduce_k(
    const float* __restrict__ kvs_part, float* __restrict__ kvs) {
  int idx = blockIdx.x * 256 + threadIdx.x;  // < 32*1920
  if (idx >= 32 * MM * DD) return;
  int hk = idx / (MM * DD), r = idx % (MM * DD);
  float s = 0.f;
  for (int c = 0; c < NCHUNK; ++c)
    s += kvs_part[((size_t)hk * NCHUNK + c) * (MM * DD) + r];
  kvs[idx] = s;
}
__global__ void __launch_bounds__(256) ks_reduce_k(
    const float* __restrict__ ks_part, float* __restrict__ ks_sum) {
  int idx = blockIdx.x * 256 + threadIdx.x;  // < 32*30
  if (idx >= 32 * MM) return;
  int hk = idx / MM, m = idx % MM;
  float s = 0.f;
  for (int c = 0; c < NCHUNK; ++c) s += ks_part[((size_t)hk * NCHUNK + c) * MM + m];
  ks_sum[idx] = s;
}

// ============ K7: z_out[n,h,d] = mean_k ( (qp @ kvs[h,k]) / (qp . ks_sum[h,k]) )
// grid = (NN/16, HH), block = 256: wave w handles gumbel sample k=w via WMMA
__global__ void __launch_bounds__(256) zout_k(
    const float* __restrict__ qp, const float* __restrict__ kvs,
    const float* __restrict__ ks_sum, float* __restrict__ zo) {
  __shared__ float qpt[16][33];           // qp tile padded M 30->32
  __shared__ float accs[KGUM][16][DD];    // per-wave num/den tiles
  const int r0 = blockIdx.x * 16;
  const int h  = blockIdx.y;
  const int t  = threadIdx.x;
  for (int i = t; i < 16 * 32; i += 256) {
    int rr = i >> 5, m = i & 31;
    qpt[rr][m] = (m < MM) ? qp[(size_t)(r0 + rr) * HM + h * MM + m] : 0.f;
  }
  __syncthreads();
  const int wave = t >> 5, lane = t & 31, l = lane & 15, half = lane >> 4;
  const int kg = wave;
  const float* kss = ks_sum + (h * KGUM + kg) * MM;
  float den[8];
#pragma unroll
  for (int i = 0; i < 8; ++i) {
    int row = i + 8 * half;
    float s = 0.f;
    for (int m = 0; m < MM; ++m) s += qpt[row][m] * kss[m];
    den[i] = s;
  }
  const float* kv = kvs + (size_t)(h * KGUM + kg) * (MM * DD);
#pragma unroll
  for (int ct = 0; ct < 4; ++ct) {
    v8f c = {0.f, 0.f, 0.f, 0.f, 0.f, 0.f, 0.f, 0.f};
    const int col = ct * 16 + l;
#pragma unroll 4
    for (int kk = 0; kk < 32; kk += 4) {
      v2f a, b;
      a.x = qpt[l][kk + 2 * half];
      a.y = qpt[l][kk + 2 * half + 1];
      const int m0 = kk + 2 * half;
      b.x = (m0 < MM) ? kv[m0 * DD + col] : 0.f;
      b.y = (m0 + 1 < MM) ? kv[(m0 + 1) * DD + col] : 0.f;
      c = wmma4(a, b, c);
    }
#pragma unroll
    for (int i = 0; i < 8; ++i) accs[wave][i + 8 * half][col] = c[i] / den[i];
  }
  __syncthreads();
  for (int i = t; i < 16 * DD; i += 256) {
    int row = i >> 6, col = i & 63;
    float s = 0.f;
#pragma unroll
    for (int w = 0; w < KGUM; ++w) s += accs[w][row][col];
    zo[(size_t)(r0 + row) * QKV_DIM + h * DD + col] = s * (1.0f / KGUM);
  }
}

// ============ K8: output projection out = zo @ Wo^T + bo ============
// grid = NN/16, block = 128 (4 waves, one 16-col tile each)
__global__ void __launch_bounds__(128) outproj_k(
    const float* __restrict__ zo, const float* __restrict__ Wo,
    const float* __restrict__ bo, float* __restrict__ out) {
  __shared__ float zt[16][260];
  const int r0 = blockIdx.x * 16;
  // async stage: 16x256 floats = 1024 x 16B chunks, 8 per thread
#pragma unroll
  for (int it = 0; it < 8; ++it) {
    int c = threadIdx.x + it * 128;             // [0,1024)
    int rr = c >> 6, cc = (c & 63) * 4;
    async_cp16(zo + (size_t)(r0 + rr) * QKV_DIM + cc, &zt[rr][cc]);
  }
  wait_async0();
  __syncthreads();
  const int wave = threadIdx.x >> 5, lane = threadIdx.x & 31;
  const int l = lane & 15, half = lane >> 4;
  const int c0 = wave * 16;
  const float* wrow = Wo + (size_t)(c0 + l) * QKV_DIM;
  v8f c = {0.f, 0.f, 0.f, 0.f, 0.f, 0.f, 0.f, 0.f};
#pragma unroll 8
  for (int kk = 0; kk < QKV_DIM; kk += 4) {
    v2f a, b;
    a.x = zt[l][kk + 2 * half];
    a.y = zt[l][kk + 2 * half + 1];
    b.x = wrow[kk + 2 * half];
    b.y = wrow[kk + 2 * half + 1];
    c = wmma4(a, b, c);
  }
  const float bb = bo[c0 + l];
#pragma unroll
  for (int i = 0; i < 8; ++i)
    out[(size_t)(r0 + i + 8 * half) * DD + c0 + l] = c[i] + bb;
}

// ============ K9: in-degrees (exact uint atomics -> deterministic) ============
__global__ void __launch_bounds__(256) din_k(const int* __restrict__ ei,
                                             unsigned* __restrict__ din) {
  int e = blockIdx.x * 256 + threadIdx.x;
  if (e < EE) atomicAdd(&din[ei[EE + e]], 1u);
}

// ============ K10: normalizer[n,h] = qp[n,h,:] . ks_tot[h,:] ============
__global__ void __launch_bounds__(256) norm_k(const float* __restrict__ qp,
                                              const float* __restrict__ ks_tot,
                                              float* __restrict__ nrm) {
  __shared__ float kt[HM];
  for (int i = threadIdx.x; i < HM; i += 256) kt[i] = ks_tot[i];
  __syncthreads();
  int idx = blockIdx.x * 256 + threadIdx.x;
  if (idx >= NN * HH) return;
  int n = idx >> 2, h = idx & 3;
  const float* qr = qp + (size_t)n * HM + h * MM;
  float s = 0.f;
  for (int m = 0; m < MM; ++m) s += qr[m] * kt[h * MM + m];
  nrm[idx] = s;
}

// ============ K11: per-edge log(A)*d_norm, block partials ============
__global__ void __launch_bounds__(256) loss_k(
    const float* __restrict__ qp, const float* __restrict__ kp,
    const float* __restrict__ nrm, const unsigned* __restrict__ din,
    const int* __restrict__ ei, float* __restrict__ part) {
  int e = blockIdx.x * 256 + threadIdx.x;
  float val = 0.f;
  if (e < EE) {
    const int s0 = ei[e], en = ei[EE + e];
    const float* qr = qp + (size_t)en * HM;
    const float* kr = kp + (size_t)s0 * HM;
    float lsum = 0.f;
#pragma unroll
    for (int h = 0; h < HH; ++h) {
      float num = 0.f;
      for (int m = 0; m < MM; ++m) num += qr[h * MM + m] * kr[h * MM + m];
      lsum += logf(num / nrm[en * HH + h]);
    }
    val = lsum / (float)din[en];
  }
  __shared__ float red[256];
  red[threadIdx.x] = val;
  __syncthreads();
  for (int st = 128; st > 0; st >>= 1) {
    if (threadIdx.x < st) red[threadIdx.x] += red[threadIdx.x + st];
    __syncthreads();
  }
  if (threadIdx.x == 0) part[blockIdx.x] = red[0];
}

// ============ K12: final scalar ============
__global__ void __launch_bounds__(256) loss_final_k(
    const float* __restrict__ part, int nparts, float* __restrict__ out) {
  float s = 0.f;
  for (int i = threadIdx.x; i < nparts; i += 256) s += part[i];
  __shared__ float red[256];
  red[threadIdx.x] = s;
  __syncthreads();
  for (int st = 128; st > 0; st >>= 1) {
    if (threadIdx.x < st) red[threadIdx.x] += red[threadIdx.x + st];
    __syncthreads();
  }
  if (threadIdx.x == 0) out[0] = red[0] * (1.0f / ((float)EE * (float)HH));
}

// =============================== launcher ===============================
extern "C" void kernel_launch(void* const* d_in, const int* in_sizes, int n_in,
                              void* d_out, int out_size, void* d_ws,
                              size_t ws_size, hipStream_t stream) {
  const float* z    = (const float*)d_in[0];
  const float* Wq   = (const float*)d_in[1];
  const float* bq   = (const float*)d_in[2];
  const float* Wk   = (const float*)d_in[3];
  const float* bk   = (const float*)d_in[4];
  const float* Wv   = (const float*)d_in[5];
  const float* bv   = (const float*)d_in[6];
  const float* Wo   = (const float*)d_in[7];
  const float* bo   = (const float*)d_in[8];
  const float* proj = (const float*)d_in[9];
  const float* gum  = (const float*)d_in[10];
  const int*   ei   = (const int*)d_in[11];
  float* out = (float*)d_out;

  // workspace layout (floats)
  float* ws = (float*)d_ws;
  size_t off = 0;
  float* q       = ws + off; off += (size_t)NN * QKV_DIM;
  float* kmat    = ws + off; off += (size_t)NN * QKV_DIM;
  float* v       = ws + off; off += (size_t)NN * QKV_DIM;
  float* zo      = ws + off; off += (size_t)NN * QKV_DIM;
  float* qp      = ws + off; off += (size_t)NN * HM;
  float* kp      = ws + off; off += (size_t)NN * HM;
  float* kvs     = ws + off; off += (size_t)HH * KGUM * MM * DD;
  float* ks_sum  = ws + off; off += (size_t)HH * KGUM * MM;
  float* ks_tot  = ws + off; off += (size_t)HM;
  float* nrm     = ws + off; off += (size_t)NN * HH;
  float* kvsp    = ws + off; off += (size_t)32 * NCHUNK * MM * DD;
  float* ksp     = ws + off; off += (size_t)32 * NCHUNK * MM;
  const int NLOSSBLK = (EE + 255) / 256;     // 1954
  float* lpart   = ws + off; off += (size_t)NLOSSBLK;
  unsigned* hmax = (unsigned*)(ws + off); off += 4;
  unsigned* din  = (unsigned*)(ws + off); off += NN;

  // reset atomic regions (graph-capture-safe async memset)
  (void)hipMemsetAsync(hmax, 0, 4 * sizeof(unsigned), stream);
  (void)hipMemsetAsync(din, 0, (size_t)NN * sizeof(unsigned), stream);

  const int ROWTILES = NN / 16;                   // 3125 (exact)
  const int NHBLK = (NN * HH + 255) / 256;        // 782

  gemm_qkv_k<<<ROWTILES, 256, 0, stream>>>(z, Wq, bq, Wk, bk, Wv, bv,
                                           q, kmat, v);
  qp_feat_k<<<NHBLK, 256, 0, stream>>>(q, proj, qp);
  kmax_k<<<NHBLK, 256, 0, stream>>>(kmat, proj, hmax);
  kp_feat_k<<<NHBLK, 256, 0, stream>>>(kmat, proj, hmax, kp);
  kstot_k<<<HM, 256, 0, stream>>>(kp, ks_tot);
  kvs_part_k<<<dim3(32, NCHUNK), 256, 0, stream>>>(kp, v, gum, kvsp, ksp);
  kvs_reduce_k<<<(32 * MM * DD + 255) / 256, 256, 0, stream>>>(kvsp, kvs);
  ks_reduce_k<<<(32 * MM + 255) / 256, 256, 0, stream>>>(ksp, ks_sum);
  zout_k<<<dim3(ROWTILES, HH), 256, 0, stream>>>(qp, kvs, ks_sum, zo);
  outproj_k<<<ROWTILES, 128, 0, stream>>>(zo, Wo, bo, out);
  din_k<<<NLOSSBLK, 256, 0, stream>>>(ei, din);
  norm_k<<<NHBLK, 256, 0, stream>>>(qp, ks_tot, nrm);
  loss_k<<<NLOSSBLK, 256, 0, stream>>>(qp, kp, nrm, din, ei, lpart);
  loss_final_k<<<1, 256, 0, stream>>>(lpart, NLOSSBLK, out + (size_t)NN * DD);
}